// DataAwareFGCN_17540646437727
// MI455X (gfx1250) — compile-verified
//
#include <hip/hip_runtime.h>
#include <hip/hip_bf16.h>

typedef __attribute__((ext_vector_type(2))) float v2f;
typedef __attribute__((ext_vector_type(8))) float v8f;

#define N_NODES 50000

// ---------------------------------------------------------------------------
// Utility: grid-stride zero fill
// ---------------------------------------------------------------------------
__global__ void fgcn_zero_kernel(float* __restrict__ p, long long n) {
    long long i = (long long)blockIdx.x * blockDim.x + threadIdx.x;
    long long stride = (long long)gridDim.x * blockDim.x;
    for (; i < n; i += stride) p[i] = 0.0f;
}

// ---------------------------------------------------------------------------
// Degree: deg[dst] += 1 per edge (self-loop +1 folded into rsqrt kernel)
// ---------------------------------------------------------------------------
__global__ void fgcn_degree_kernel(const int* __restrict__ dst, float* __restrict__ deg, int E) {
    int e = blockIdx.x * blockDim.x + threadIdx.x;
    if (e < E) atomicAdd(&deg[dst[e]], 1.0f);
}

__global__ void fgcn_dinv_kernel(float* __restrict__ deg_dinv, int n) {
    int i = blockIdx.x * blockDim.x + threadIdx.x;
    if (i < n) deg_dinv[i] = rsqrtf(deg_dinv[i] + 1.0f);  // +1 = self loop; deg>=1 so no clamp
}

// ---------------------------------------------------------------------------
// Dense GEMM via fp32 WMMA: C[M,NOUT] = A[M,K] @ W[K,NOUT]
// One 16x16 output tile per wave; K-loop of 16x16x4 fp32 WMMAs.
// M must be a multiple of 16 (50000 = 3125*16).
// ---------------------------------------------------------------------------
template <int K, int NOUT>
__global__ void fgcn_wmma_gemm_kernel(const float* __restrict__ A,
                                      const float* __restrict__ W,
                                      float* __restrict__ C, int M) {
    const int wave = threadIdx.x >> 5;
    const int lane = threadIdx.x & 31;
    const int ntiles = NOUT / 16;
    const int tile = blockIdx.x * (blockDim.x >> 5) + wave;
    const int total_tiles = (M / 16) * ntiles;
    if (tile >= total_tiles) return;               // wave-uniform: EXEC stays all-ones

    const int m0 = (tile / ntiles) * 16;
    const int n0 = (tile % ntiles) * 16;
    const int row   = lane & 15;                   // M (for A) / N (for B, C/D)
    const int half  = lane >> 4;                   // K-half selector
    const int kbase = half * 2;

    v8f acc = {0.f, 0.f, 0.f, 0.f, 0.f, 0.f, 0.f, 0.f};

    const float* arow = A + (size_t)(m0 + row) * K + kbase;         // A[m0+row][kbase..]
    const float* wcol = W + (size_t)kbase * NOUT + (n0 + row);      // W[kbase][n0+row]

#pragma unroll 4
    for (int k0 = 0; k0 < K; k0 += 4) {
        v2f a, b;
        a.x = arow[0];                 // A[m][k0+kbase+0]
        a.y = arow[1];                 // A[m][k0+kbase+1]
        b.x = wcol[0];                 // W[k0+kbase+0][n]
        b.y = wcol[NOUT];              // W[k0+kbase+1][n]
        acc = __builtin_amdgcn_wmma_f32_16x16x4_f32(false, a, false, b,
                                                    (short)0, acc, false, false);
        arow += 4;
        wcol += 4 * NOUT;
    }

    // C/D layout: VGPR i -> M = i + 8*half; N = lane&15
    const int mbase = m0 + half * 8;
    const int col = n0 + row;
#pragma unroll
    for (int i = 0; i < 8; ++i)
        C[(size_t)(mbase + i) * NOUT + col] = acc[i];
}

// ---------------------------------------------------------------------------
// Edge scatter: agg[dst] += dinv[src]*dinv[dst] * h[src]
// Consecutive lanes cover consecutive features of one edge -> coalesced
// gathers and coalesced global_atomic_add_f32.
// ---------------------------------------------------------------------------
template <int D>
__global__ void fgcn_scatter_kernel(const float* __restrict__ h,
                                    const int* __restrict__ src,
                                    const int* __restrict__ dst,
                                    const float* __restrict__ dinv,
                                    float* __restrict__ agg, long long total) {
    long long idx = (long long)blockIdx.x * blockDim.x + threadIdx.x;
    if (idx >= total) return;
    int e = (int)(idx / D);
    int f = (int)(idx % D);
    int s = src[e];
    int d = dst[e];
    float nrm = dinv[s] * dinv[d];
    atomicAdd(&agg[(size_t)d * D + f], nrm * h[(size_t)s * D + f]);
}

// ---------------------------------------------------------------------------
// h = relu(agg + dinv^2 * hx + b)   (self-loop contribution folded in)
// In-place on the agg buffer.
// ---------------------------------------------------------------------------
template <int D>
__global__ void fgcn_relu_bias_self_kernel(const float* __restrict__ hx,
                                           const float* __restrict__ bias,
                                           const float* __restrict__ dinv,
                                           float* __restrict__ agg_h, int n) {
    long long idx = (long long)blockIdx.x * blockDim.x + threadIdx.x;
    long long total = (long long)n * D;
    if (idx >= total) return;
    int node = (int)(idx / D);
    int f = (int)(idx % D);
    float di = dinv[node];
    float v = agg_h[idx] + di * di * hx[idx] + bias[f];
    agg_h[idx] = fmaxf(v, 0.0f);
}

// ---------------------------------------------------------------------------
// out = h2 * sigmoid(h2 @ attn_w + attn_b)  — one wave32 per node (32 feats)
// ---------------------------------------------------------------------------
__global__ void fgcn_attn_out_kernel(const float* __restrict__ h2,
                                     const float* __restrict__ aw,
                                     const float* __restrict__ ab,
                                     float* __restrict__ out, int n) {
    int node = blockIdx.x * (blockDim.x >> 5) + (threadIdx.x >> 5);
    int lane = threadIdx.x & 31;
    if (node >= n) return;
    float v = h2[(size_t)node * 32 + lane];
    float p = v * aw[lane];
#pragma unroll
    for (int m = 16; m; m >>= 1) p += __shfl_xor(p, m, 32);
    float a = 1.0f / (1.0f + __expf(-(p + ab[0])));
    out[(size_t)node * 32 + lane] = v * a;
}

// ---------------------------------------------------------------------------
extern "C" void kernel_launch(void* const* d_in, const int* in_sizes, int n_in,
                              void* d_out, int out_size, void* d_ws, size_t ws_size,
                              hipStream_t stream) {
    const float* x  = (const float*)d_in[0];   // [N,128]
    const int*   ei = (const int*)d_in[1];     // [2,E]
    const float* W1 = (const float*)d_in[2];   // [128,64]
    const float* b1 = (const float*)d_in[3];   // [64]
    const float* W2 = (const float*)d_in[4];   // [64,32]
    const float* b2 = (const float*)d_in[5];   // [32]
    const float* aw = (const float*)d_in[6];   // [32,1]
    const float* ab = (const float*)d_in[7];   // [1]

    const int N = N_NODES;
    const int E = in_sizes[1] / 2;             // 800000
    const int* src = ei;                       // edge_index[0]
    const int* dst = ei + E;                   // edge_index[1]

    // Workspace layout (floats)
    float* ws   = (float*)d_ws;
    float* dinv = ws;                          // [N]    (deg, then dinv in place)
    float* hx1  = ws + 50048;                  // [N,64] x @ W1
    float* h1   = hx1 + (size_t)N * 64;        // [N,64] agg1, then relu'd h1 in place
    float* hx2  = h1 + (size_t)N * 64;         // [N,32] h1 @ W2
    float* h2   = hx2 + (size_t)N * 32;        // [N,32] agg2, then relu'd h2 in place

    const int B = 256;

    // 1) zero accumulators
    fgcn_zero_kernel<<<1024, B, 0, stream>>>(dinv, (long long)N);
    fgcn_zero_kernel<<<4096, B, 0, stream>>>(h1, (long long)N * 64);
    fgcn_zero_kernel<<<4096, B, 0, stream>>>(h2, (long long)N * 32);

    // 2) degree + symmetric normalization
    fgcn_degree_kernel<<<(E + B - 1) / B, B, 0, stream>>>(dst, dinv, E);
    fgcn_dinv_kernel<<<(N + B - 1) / B, B, 0, stream>>>(dinv, N);

    // 3) layer 1: GEMM (WMMA f32) -> scatter -> relu+bias+self-loop
    {
        int tiles = (N / 16) * (64 / 16);                 // 12500
        int blocks = (tiles + (B / 32) - 1) / (B / 32);
        fgcn_wmma_gemm_kernel<128, 64><<<blocks, B, 0, stream>>>(x, W1, hx1, N);
        long long total = (long long)E * 64;
        fgcn_scatter_kernel<64><<<(int)((total + B - 1) / B), B, 0, stream>>>(
            hx1, src, dst, dinv, h1, total);
        long long nt = (long long)N * 64;
        fgcn_relu_bias_self_kernel<64><<<(int)((nt + B - 1) / B), B, 0, stream>>>(
            hx1, b1, dinv, h1, N);
    }

    // 4) layer 2
    {
        int tiles = (N / 16) * (32 / 16);                 // 6250
        int blocks = (tiles + (B / 32) - 1) / (B / 32);
        fgcn_wmma_gemm_kernel<64, 32><<<blocks, B, 0, stream>>>(h1, W2, hx2, N);
        long long total = (long long)E * 32;
        fgcn_scatter_kernel<32><<<(int)((total + B - 1) / B), B, 0, stream>>>(
            hx2, src, dst, dinv, h2, total);
        long long nt = (long long)N * 32;
        fgcn_relu_bias_self_kernel<32><<<(int)((nt + B - 1) / B), B, 0, stream>>>(
            hx2, b2, dinv, h2, N);
    }

    // 5) attention gate + output
    {
        int waves_per_block = B / 32;
        int blocks = (N + waves_per_block - 1) / waves_per_block;
        fgcn_attn_out_kernel<<<blocks, B, 0, stream>>>(h2, aw, ab, (float*)d_out, N);
    }
}